// HyperNetwork_66640712565459
// MI455X (gfx1250) — compile-verified
//
#include <hip/hip_runtime.h>
#include <hip/hip_bf16.h>

typedef __attribute__((ext_vector_type(2))) float v2f;
typedef __attribute__((ext_vector_type(4))) float v4f;
typedef __attribute__((ext_vector_type(8))) float v8f;

#define IN_CH  1024
#define Z_DIM  128
#define NZ     1024
#define OUT_K  72          // OUT_CH * F*F = 8*9
#define F2     9

// ---------------------------------------------------------------------------
// Stage 1: batched per-channel GEMV  a[c,d] = W_in[c,d,:] . x[c,:] + b_in[c,d]
// Pure HBM stream over 536 MB of W_in -> non-temporal b128 loads, wave32
// shuffle reduction. One block (8 waves) per channel, 16 rows per wave.
// ---------------------------------------------------------------------------
__global__ __launch_bounds__(256) void hyper_gemv(const float* __restrict__ z,
                                                  const float* __restrict__ W_in,
                                                  const float* __restrict__ b_in,
                                                  float* __restrict__ a_ws) {
    const int c    = blockIdx.x;
    const int tid  = threadIdx.x;
    const int wave = tid >> 5;     // 0..7
    const int lane = tid & 31;     // wave32

    // Pre-load this lane's slice of x[c] (same slice reused for every row):
    // lane covers elements n = (i*32 + lane)*4 .. +3, i = 0..7  (32 floats)
    const v4f* xv = (const v4f*)(z + (size_t)c * NZ);
    v4f xr[8];
#pragma unroll
    for (int i = 0; i < 8; ++i) xr[i] = xv[i * 32 + lane];

    const float* Wc = W_in + (size_t)c * Z_DIM * NZ;

#pragma unroll 1
    for (int r = 0; r < 16; ++r) {
        const int d = wave * 16 + r;
        const v4f* wrow = (const v4f*)(Wc + d * NZ);
        float acc = 0.0f;
#pragma unroll
        for (int i = 0; i < 8; ++i) {
            // coalesced: 32 lanes x 16B = 512B contiguous per instruction;
            // non-temporal: W_in is read exactly once, keep it out of caches
            v4f w4 = __builtin_nontemporal_load(&wrow[i * 32 + lane]);
            v4f x4 = xr[i];
            acc += w4.x * x4.x;
            acc += w4.y * x4.y;
            acc += w4.z * x4.z;
            acc += w4.w * x4.w;
        }
        // wave32 butterfly reduction
#pragma unroll
        for (int off = 16; off > 0; off >>= 1)
            acc += __shfl_xor(acc, off, 32);
        if (lane == 0)
            a_ws[c * Z_DIM + d] = acc + b_in[c * Z_DIM + d];
    }
}

// ---------------------------------------------------------------------------
// Stage 2: k = a @ W_out^T + b_out, scattered into (OUT_CH, C, 3, 3).
// Real GEMM (M=1024, N=72, K=128) -> V_WMMA_F32_16X16X4_F32.
// One wave per 16x16 C tile; 32 k-steps of 4, fully unrolled.
//
// A layout (32-bit A 16x4): lanes 0-15 -> M=lane, {K0,K1}; lanes 16-31 ->
// M=lane-16, {K2,K3}.  B assumed transpose-symmetric: lane holds
// B[K=2*half+j][N=lm] in component j.  C/D: VGPR r -> M = r + 8*half, N = lm.
// ---------------------------------------------------------------------------
__global__ __launch_bounds__(256) void hyper_wmma_out(const float* __restrict__ a_ws,
                                                      const float* __restrict__ W_out,
                                                      const float* __restrict__ b_out,
                                                      float* __restrict__ out) {
    const int lane = threadIdx.x;      // 0..31 (wave32)
    const int half = lane >> 4;        // selects K pair
    const int lm   = lane & 15;

    const int ctile = blockIdx.x * 8 + threadIdx.y;   // 0..63
    const int cbase = ctile * 16;
    const int nbase = blockIdx.y * 16;                // 0,16,32,48,64

    const int  n      = nbase + lm;
    const bool nvalid = (n < OUT_K);

    // per-lane base pointers (clamp B row so the load is always in-bounds;
    // value is select-zeroed -> EXEC stays all-1s for the WMMA)
    const float* arow = a_ws + (cbase + lm) * Z_DIM + 2 * half;
    const float* brow = W_out + (nvalid ? n : (OUT_K - 1)) * Z_DIM + 2 * half;

    v8f acc = {};
#pragma unroll
    for (int ko = 0; ko < Z_DIM; ko += 4) {
        v2f A = *(const v2f*)(arow + ko);              // a[cbase+lm][ko+2h..+1]
        v2f Braw = *(const v2f*)(brow + ko);           // W_out[n][ko+2h..+1]
        v2f B = nvalid ? Braw : (v2f)0.0f;
        acc = __builtin_amdgcn_wmma_f32_16x16x4_f32(
            /*neg_a=*/false, A, /*neg_b=*/false, B,
            /*c_mod=*/(short)0, acc, /*reuse_a=*/false, /*reuse_b=*/false);
    }

    // epilogue: +bias, scatter k[c, n] -> out[n/9][c][n%9]
    if (nvalid) {
        const float bb = b_out[n];
        const int og = n / F2;
        const int ol = n % F2;
#pragma unroll
        for (int r = 0; r < 8; ++r) {
            const int cc = cbase + r + 8 * half;
            out[og * (IN_CH * F2) + cc * F2 + ol] = acc[r] + bb;
        }
    }
}

extern "C" void kernel_launch(void* const* d_in, const int* in_sizes, int n_in,
                              void* d_out, int out_size, void* d_ws, size_t ws_size,
                              hipStream_t stream) {
    const float* z     = (const float*)d_in[0];   // (1024, 32, 32)
    const float* W_in  = (const float*)d_in[1];   // (1024, 128, 1024)
    const float* b_in  = (const float*)d_in[2];   // (1024, 128)
    const float* W_out = (const float*)d_in[3];   // (72, 128)
    const float* b_out = (const float*)d_in[4];   // (72,)
    float*       out   = (float*)d_out;           // (8, 1024, 3, 3)
    float*       a_ws  = (float*)d_ws;            // 1024*128 floats = 512 KB

    // Stage 1: one block per channel, 8 waves x 16 rows
    hyper_gemv<<<dim3(IN_CH), dim3(256), 0, stream>>>(z, W_in, b_in, a_ws);

    // Stage 2: 64 M-tiles x 5 N-tiles; 8 waves (M-tiles) per block
    hyper_wmma_out<<<dim3(8, 5), dim3(32, 8), 0, stream>>>(a_ws, W_out, b_out, out);
}